// MLP_31774168055825
// MI455X (gfx1250) — compile-verified
//
#include <hip/hip_runtime.h>
#include <math.h>

typedef __attribute__((ext_vector_type(2))) float v2f;
typedef __attribute__((ext_vector_type(8))) float v8f;

#define NPIX   (512*512)     // pixels per (sample, channel) plane
#define NSAMP  16
#define CI     3
#define CH     16
#define CO     3
#define PB1    64            // partial blocks per sample, stats kernels
#define PB3    64
#define PB5    128
#define EPS    1e-7f
#define SLOPE  0.01f

// D = A(16x4) * B(4x16) + C, all f32. 8-arg form per probe doc.
__device__ __forceinline__ v8f wmma4(v2f a, v2f b, v8f c) {
  return __builtin_amdgcn_wmma_f32_16x16x4_f32(false, a, false, b, (short)0, c,
                                               false, false);
}

// ---------------- Kernel 1: per-sample stats of y1 = Wa*x + ba ----------------
__global__ void k_stats1(const float* __restrict__ x, const float* __restrict__ Wa,
                         const float* __restrict__ ba, float* __restrict__ part) {
  const int s = blockIdx.y;
  const float* xs = x + (size_t)s * CI * NPIX;

  float wa[CH][CI], bav[CH];
#pragma unroll
  for (int c = 0; c < CH; ++c) {
    bav[c] = ba[c];
#pragma unroll
    for (int i = 0; i < CI; ++i) wa[c][i] = Wa[c * CI + i];
  }

  float s1 = 0.f, s2 = 0.f;
  const int stride = blockDim.x * gridDim.x;
  for (int p = blockIdx.x * blockDim.x + threadIdx.x; p < NPIX; p += stride) {
    float x0 = xs[p], x1 = xs[NPIX + p], x2 = xs[2 * NPIX + p];
#pragma unroll
    for (int c = 0; c < CH; ++c) {
      float y = fmaf(wa[c][0], x0, fmaf(wa[c][1], x1, fmaf(wa[c][2], x2, bav[c])));
      s1 += y;
      s2 += y * y;
    }
  }

  __shared__ float r1[256], r2[256];
  r1[threadIdx.x] = s1; r2[threadIdx.x] = s2;
  __syncthreads();
  for (int w = 128; w > 0; w >>= 1) {
    if (threadIdx.x < (unsigned)w) {
      r1[threadIdx.x] += r1[threadIdx.x + w];
      r2[threadIdx.x] += r2[threadIdx.x + w];
    }
    __syncthreads();
  }
  if (threadIdx.x == 0) {
    part[(s * PB1 + blockIdx.x) * 2 + 0] = r1[0];
    part[(s * PB1 + blockIdx.x) * 2 + 1] = r2[0];
  }
}

// ------------- Finalize: fold partials -> (mean, 1/(sqrt(var)+eps)) ----------
__global__ void k_finalize(const float* __restrict__ part, float* __restrict__ stats,
                           int nb) {
  int s = threadIdx.x;
  if (s >= NSAMP) return;
  double s1 = 0.0, s2 = 0.0;
  for (int b = 0; b < nb; ++b) {
    s1 += (double)part[(s * nb + b) * 2 + 0];
    s2 += (double)part[(s * nb + b) * 2 + 1];
  }
  const double n = (double)CH * (double)NPIX;
  double mean = s1 / n;
  double var = (s2 - s1 * mean) / (n - 1.0);
  stats[2 * s + 0] = (float)mean;
  stats[2 * s + 1] = 1.0f / (sqrtf((float)var) + EPS);
}

// --------- Kernel 3: per-sample stats of z = Wb*lrelu(norm(y1)) + bb ---------
__global__ void k_stats2(const float* __restrict__ x, const float* __restrict__ Wa,
                         const float* __restrict__ ba, const float* __restrict__ Wb,
                         const float* __restrict__ bb, const float* __restrict__ stats1,
                         float* __restrict__ part) {
  const int s = blockIdx.y;
  const float* xs = x + (size_t)s * CI * NPIX;
  const int lane  = threadIdx.x & 31;
  const int wave  = threadIdx.x >> 5;
  const int m     = lane & 15;              // row / col-in-tile index
  const int khalf = (lane >> 4) << 1;       // 0 (lanes 0-15) or 2 (lanes 16-31)
  const int hi8   = (lane >> 4) << 3;       // C/D row offset: 0 or 8

  // A operand for conv_a: Wa padded to 16x4 (K=3 column is zero)
  v2f aA;
  aA.x = Wa[m * CI + khalf];                          // K = 0 or 2
  aA.y = (khalf == 0) ? Wa[m * CI + 1] : 0.f;         // K = 1 or 3(->0)

  // A chunks for conv_b: Wb 16x16, chunk j covers K = 4j..4j+3
  v2f bA[4];
#pragma unroll
  for (int j = 0; j < 4; ++j) {
    int k0 = 4 * j + khalf;
    bA[j].x = Wb[m * CH + k0];
    bA[j].y = Wb[m * CH + k0 + 1];
  }

  v8f biasA, biasB;
#pragma unroll
  for (int i = 0; i < 8; ++i) { biasA[i] = ba[i + hi8]; biasB[i] = bb[i + hi8]; }

  const float mean1 = stats1[2 * s], inv1 = stats1[2 * s + 1];

  __shared__ float tile[8][CH * 16];   // one 16x16 f32 tile per wave
  float* tl = &tile[wave][0];

  float acc1 = 0.f, acc2 = 0.f;
  const int tpb = (NPIX / 16) / gridDim.x;  // tiles per block
  const int tpw = tpb / 8;                  // tiles per wave
  const int tbase = blockIdx.x * tpb + wave * tpw;

  for (int t = 0; t < tpw; ++t) {
    const int p = (tbase + t) * 16;
    // B operand for conv_a: x tile (4x16, channel 3 = zero)
    v2f xB;
    xB.x = xs[khalf * NPIX + p + m];
    xB.y = (khalf == 0) ? xs[NPIX + p + m] : 0.f;

    v8f y = wmma4(aA, xB, biasA);          // y1 tile, C/D layout

    v8f h;
#pragma unroll
    for (int i = 0; i < 8; ++i) {
      float v = (y[i] - mean1) * inv1;
      h[i] = (v >= 0.f) ? v : SLOPE * v;
    }

    __builtin_amdgcn_wave_barrier();
#pragma unroll
    for (int i = 0; i < 8; ++i) tl[(i + hi8) * 16 + m] = h[i];  // (ch, pix)
    __builtin_amdgcn_wave_barrier();

    v8f z = biasB;
#pragma unroll
    for (int j = 0; j < 4; ++j) {
      int k0 = 4 * j + khalf;
      v2f hB;
      hB.x = tl[k0 * 16 + m];
      hB.y = tl[(k0 + 1) * 16 + m];
      z = wmma4(bA[j], hB, z);
    }
    __builtin_amdgcn_wave_barrier();

#pragma unroll
    for (int i = 0; i < 8; ++i) { acc1 += z[i]; acc2 += z[i] * z[i]; }
  }

  __shared__ float r1[256], r2[256];
  r1[threadIdx.x] = acc1; r2[threadIdx.x] = acc2;
  __syncthreads();
  for (int w = 128; w > 0; w >>= 1) {
    if (threadIdx.x < (unsigned)w) {
      r1[threadIdx.x] += r1[threadIdx.x + w];
      r2[threadIdx.x] += r2[threadIdx.x + w];
    }
    __syncthreads();
  }
  if (threadIdx.x == 0) {
    part[(s * PB3 + blockIdx.x) * 2 + 0] = r1[0];
    part[(s * PB3 + blockIdx.x) * 2 + 1] = r2[0];
  }
}

// ------------------- Kernel 5: full fused chain -> output --------------------
__global__ void k_final(const float* __restrict__ x, const float* __restrict__ Wa,
                        const float* __restrict__ ba, const float* __restrict__ Wb,
                        const float* __restrict__ bb, const float* __restrict__ Wc,
                        const float* __restrict__ bc, const float* __restrict__ stats1,
                        const float* __restrict__ stats2, float* __restrict__ out) {
  const int s = blockIdx.y;
  const float* xs = x + (size_t)s * CI * NPIX;
  float* os = out + (size_t)s * CO * NPIX;
  const int lane  = threadIdx.x & 31;
  const int wave  = threadIdx.x >> 5;
  const int m     = lane & 15;
  const int khalf = (lane >> 4) << 1;
  const int hi8   = (lane >> 4) << 3;

  v2f aA;
  aA.x = Wa[m * CI + khalf];
  aA.y = (khalf == 0) ? Wa[m * CI + 1] : 0.f;

  v2f bA[4], cA[4];
#pragma unroll
  for (int j = 0; j < 4; ++j) {
    int k0 = 4 * j + khalf;
    bA[j].x = Wb[m * CH + k0];
    bA[j].y = Wb[m * CH + k0 + 1];
    cA[j].x = (m < CO) ? Wc[m * CH + k0] : 0.f;       // Wc padded to 16 rows
    cA[j].y = (m < CO) ? Wc[m * CH + k0 + 1] : 0.f;
  }

  v8f biasA, biasB, biasC;
#pragma unroll
  for (int i = 0; i < 8; ++i) {
    biasA[i] = ba[i + hi8];
    biasB[i] = bb[i + hi8];
    biasC[i] = (i + hi8 < CO) ? bc[i + hi8] : 0.f;
  }

  const float mean1 = stats1[2 * s], inv1 = stats1[2 * s + 1];
  const float mean2 = stats2[2 * s], inv2 = stats2[2 * s + 1];

  __shared__ float tile[8][CH * 16];
  float* tl = &tile[wave][0];

  const int tpb = (NPIX / 16) / gridDim.x;
  const int tpw = tpb / 8;
  const int tbase = blockIdx.x * tpb + wave * tpw;

  for (int t = 0; t < tpw; ++t) {
    const int p = (tbase + t) * 16;
    v2f xB;
    xB.x = xs[khalf * NPIX + p + m];
    xB.y = (khalf == 0) ? xs[NPIX + p + m] : 0.f;

    v8f y = wmma4(aA, xB, biasA);

    v8f h1;
#pragma unroll
    for (int i = 0; i < 8; ++i) {
      float v = (y[i] - mean1) * inv1;
      h1[i] = (v >= 0.f) ? v : SLOPE * v;
    }

    __builtin_amdgcn_wave_barrier();
#pragma unroll
    for (int i = 0; i < 8; ++i) tl[(i + hi8) * 16 + m] = h1[i];
    __builtin_amdgcn_wave_barrier();

    v8f z = biasB;
#pragma unroll
    for (int j = 0; j < 4; ++j) {
      int k0 = 4 * j + khalf;
      v2f hB;
      hB.x = tl[k0 * 16 + m];
      hB.y = tl[(k0 + 1) * 16 + m];
      z = wmma4(bA[j], hB, z);
    }
    __builtin_amdgcn_wave_barrier();

    v8f h2;
#pragma unroll
    for (int i = 0; i < 8; ++i) {
      float v = (z[i] - mean2) * inv2;
      h2[i] = (v >= 0.f) ? v : SLOPE * v;
    }

    __builtin_amdgcn_wave_barrier();
#pragma unroll
    for (int i = 0; i < 8; ++i) tl[(i + hi8) * 16 + m] = h2[i];
    __builtin_amdgcn_wave_barrier();

    v8f o = biasC;
#pragma unroll
    for (int j = 0; j < 4; ++j) {
      int k0 = 4 * j + khalf;
      v2f hB;
      hB.x = tl[k0 * 16 + m];
      hB.y = tl[(k0 + 1) * 16 + m];
      o = wmma4(cA[j], hB, o);
    }
    __builtin_amdgcn_wave_barrier();

    if (lane < 16) {                      // rows 0..2 live in lanes 0-15
      os[0 * NPIX + p + m] = o[0];
      os[1 * NPIX + p + m] = o[1];
      os[2 * NPIX + p + m] = o[2];
    }
  }
}

extern "C" void kernel_launch(void* const* d_in, const int* in_sizes, int n_in,
                              void* d_out, int out_size, void* d_ws, size_t ws_size,
                              hipStream_t stream) {
  (void)in_sizes; (void)n_in; (void)out_size; (void)ws_size;
  const float* x  = (const float*)d_in[0];
  const float* Wa = (const float*)d_in[1];
  const float* ba = (const float*)d_in[2];
  const float* Wb = (const float*)d_in[3];
  const float* bb = (const float*)d_in[4];
  const float* Wc = (const float*)d_in[5];
  const float* bc = (const float*)d_in[6];
  float* out = (float*)d_out;

  float* ws     = (float*)d_ws;
  float* part1  = ws;                                  // 16*64*2 = 2048 f
  float* stats1 = part1 + NSAMP * PB1 * 2;             // 32 f
  float* part2  = stats1 + NSAMP * 2;                  // 2048 f
  float* stats2 = part2 + NSAMP * PB3 * 2;             // 32 f

  k_stats1 <<<dim3(PB1, NSAMP), 256, 0, stream>>>(x, Wa, ba, part1);
  k_finalize<<<1, 32, 0, stream>>>(part1, stats1, PB1);
  k_stats2 <<<dim3(PB3, NSAMP), 256, 0, stream>>>(x, Wa, ba, Wb, bb, stats1, part2);
  k_finalize<<<1, 32, 0, stream>>>(part2, stats2, PB3);
  k_final  <<<dim3(PB5, NSAMP), 256, 0, stream>>>(x, Wa, ba, Wb, bb, Wc, bc,
                                                  stats1, stats2, out);
}